// EmformerAttention_4698694222676
// MI455X (gfx1250) — compile-verified
//
#include <hip/hip_runtime.h>
#include <hip/hip_bf16.h>
#include <math.h>

// Problem constants
#define Tn   1024
#define Bn   4
#define Dn   1024
#define Rn   64
#define Sn   16
#define Mn   64
#define Hn   16
#define DHn  64
#define Qn   1104   // R+T+S
#define KLn  1152   // M+R+T
#define QBn  4416   // Q*B
#define KLBn 4608   // KL*B
#define QSn  1088   // Q-S
#define NEG_INF (-100000000.0f)
#define SCALING 0.125f  // dh^-0.5 = 1/8

typedef __attribute__((ext_vector_type(2))) float v2f;
typedef __attribute__((ext_vector_type(8))) float v8f;
typedef __attribute__((ext_vector_type(4))) unsigned int v4u;
typedef __attribute__((ext_vector_type(4))) int v4i;
typedef __attribute__((ext_vector_type(8))) int v8i;

#if __has_builtin(__builtin_amdgcn_tensor_load_to_lds)
#define HAVE_TDM 1
#else
#define HAVE_TDM 0
#endif

__device__ __forceinline__ v8f vzero8() {
  v8f z;
#pragma unroll
  for (int i = 0; i < 8; ++i) z[i] = 0.0f;
  return z;
}

// f32 WMMA: D(16x16) = A(16x4) * B(4x16) + C
__device__ __forceinline__ v8f wmma4(v2f a, v2f b, v8f c) {
  return __builtin_amdgcn_wmma_f32_16x16x4_f32(
      false, a, false, b, (short)0, c, false, false);
}

#if HAVE_TDM
// Generic pointer to an LDS object: low 32 bits are the LDS byte offset.
__device__ __forceinline__ unsigned lds_addr_of(const void* p) {
  return (unsigned)(unsigned long long)p;
}

// One 2-D TDM tile load: tileW f32 elems x tileH rows, row stride in elems;
// LDS row padding via D# pad fields. D# bitfields per CDNA5 ISA 8.3-8.6.
__device__ __forceinline__ void tdm_load_2d(const void* gsrc, unsigned ldsByteOff,
                                            unsigned tileW, unsigned tileH,
                                            unsigned rowStrideElems,
                                            unsigned padIntervalCode,
                                            unsigned padAmountCode,
                                            bool padEnable) {
  unsigned long long ga = (unsigned long long)gsrc;
  v4u g0;
  g0[0] = 1u;                                      // count=1, user mode
  g0[1] = ldsByteOff;                              // lds_addr (bytes)
  g0[2] = (unsigned)ga;                            // global_addr[31:0]
  g0[3] = (unsigned)((ga >> 32) & 0x01FFFFFFu) | 0x80000000u;  // hi + type=2
  unsigned flags = (2u << 16);                     // data_size = 4B
  if (padEnable)
    flags |= (1u << 20) | (padIntervalCode << 22) | (padAmountCode << 25);
  v8i g1;
  g1[0] = (int)flags;
  g1[1] = (int)((tileW & 0xFFFFu) << 16);          // tensor_dim0 lo
  g1[2] = (int)(((tileW >> 16) & 0xFFFFu) | ((tileH & 0xFFFFu) << 16));
  g1[3] = (int)(((tileH >> 16) & 0xFFFFu) | ((tileW & 0xFFFFu) << 16));
  g1[4] = (int)(tileH & 0xFFFFu);                  // tile_dim1, tile_dim2=0
  g1[5] = (int)rowStrideElems;                     // tensor_dim0_stride lo
  g1[6] = 0;
  g1[7] = 0;
  v4i g2 = {0, 0, 0, 0};
  v4i g3 = {0, 0, 0, 0};
#if __clang_major__ >= 23
  v8i g4 = {0, 0, 0, 0, 0, 0, 0, 0};
  __builtin_amdgcn_tensor_load_to_lds(g0, g1, g2, g3, g4, 0);
#else
  __builtin_amdgcn_tensor_load_to_lds(g0, g1, g2, g3, 0);
#endif
}
#endif  // HAVE_TDM

// Row gather across the 3 concatenated source segments (each row is Dn floats)
__device__ __forceinline__ const float* gather_row(const float* s0, int n0,
                                                   const float* s1, int n1,
                                                   const float* s2, int r) {
  if (r < n0) return s0 + (size_t)r * Dn;
  r -= n0;
  if (r < n1) return s1 + (size_t)r * Dn;
  return s2 + (size_t)(r - n1) * Dn;
}

// ---------------------------------------------------------------------------
// Projection GEMM: out[r, c] = concat(seg0,seg1,seg2)[r, :] @ W[:, c] + bias[c]
// 64x128 block tile, 8 waves. X tile: gathered vector loads.
// W tiles: TDM, double-buffered (chunk i+1 streams in behind compute on i).
// ---------------------------------------------------------------------------
__global__ __launch_bounds__(256) void proj_gemm(
    const float* __restrict__ seg0, int n0,
    const float* __restrict__ seg1, int n1,
    const float* __restrict__ seg2,
    const float* __restrict__ W, const float* __restrict__ bias, int Dout,
    float* __restrict__ out0, float* __restrict__ out1) {
  extern __shared__ float sm[];
  float* Xs  = sm;               // 64 x 68
  float* Wb0 = Xs + 64 * 68;     // 64 x 132
  float* Wb1 = Wb0 + 64 * 132;   // 64 x 132

  const int tid = threadIdx.x;
  const int w = tid >> 5, lane = tid & 31;
  const int hf = lane >> 4, l = lane & 15;
  const int rt = w >> 1, cg = w & 1;
  const int row0 = blockIdx.x * 64;
  const int nbase = blockIdx.y * 128;
  const int NCH = Dn / 64;       // 16 K-chunks

  v8f acc[4];
#pragma unroll
  for (int j = 0; j < 4; ++j) acc[j] = vzero8();

#if HAVE_TDM
  if (w == 0)  // prologue: chunk 0 -> Wb0
    tdm_load_2d(W + nbase, lds_addr_of(Wb0), 128, 64, (unsigned)Dout, 6, 3, true);
#endif

  for (int ic = 0; ic < NCH; ++ic) {
    const int k0 = ic * 64;
    __syncthreads();  // compute of ic-1 done: safe to overwrite Xs / other Wb
    // Stage X tile (64x64): 1024 float4s, 4 per thread
#pragma unroll
    for (int j = 0; j < 4; ++j) {
      int vi = tid + j * 256;
      int row = vi >> 4, c = (vi & 15) << 2;
      const float* src = gather_row(seg0, n0, seg1, n1, seg2, row0 + row) + k0 + c;
      float4 v = *(const float4*)src;
      Xs[row * 68 + c + 0] = v.x; Xs[row * 68 + c + 1] = v.y;
      Xs[row * 68 + c + 2] = v.z; Xs[row * 68 + c + 3] = v.w;
    }
#if HAVE_TDM
    if (w == 0) {
      if (ic + 1 < NCH) {
        tdm_load_2d(W + (size_t)(k0 + 64) * Dout + nbase,
                    lds_addr_of((ic & 1) ? Wb0 : Wb1),
                    128, 64, (unsigned)Dout, 6, 3, true);
        __builtin_amdgcn_s_wait_tensorcnt(1);  // chunk ic landed; ic+1 in flight
      } else {
        __builtin_amdgcn_s_wait_tensorcnt(0);
      }
    }
#else
    {
      float* Wst = (ic & 1) ? Wb1 : Wb0;
#pragma unroll
      for (int j = 0; j < 8; ++j) {
        int vi = tid + j * 256;
        int row = vi >> 5, c = (vi & 31) << 2;
        float4 v = *(const float4*)(W + (size_t)(k0 + row) * Dout + nbase + c);
        Wst[row * 132 + c + 0] = v.x; Wst[row * 132 + c + 1] = v.y;
        Wst[row * 132 + c + 2] = v.z; Wst[row * 132 + c + 3] = v.w;
      }
    }
#endif
    __syncthreads();

    const float* Wc = (ic & 1) ? Wb1 : Wb0;
#pragma unroll 4
    for (int kk = 0; kk < 64; kk += 4) {
      v2f a;
      a.x = Xs[(rt * 16 + l) * 68 + kk + 2 * hf];
      a.y = Xs[(rt * 16 + l) * 68 + kk + 2 * hf + 1];
#pragma unroll
      for (int j = 0; j < 4; ++j) {
        int nt = cg * 64 + j * 16;
        v2f b;
        b.x = Wc[(kk + 2 * hf) * 132 + nt + l];
        b.y = Wc[(kk + 2 * hf + 1) * 132 + nt + l];
        acc[j] = wmma4(a, b, acc[j]);
      }
    }
  }

#pragma unroll
  for (int j = 0; j < 4; ++j) {
    int col = nbase + cg * 64 + j * 16 + l;
    float bv = bias[col];
#pragma unroll
    for (int i = 0; i < 8; ++i) {
      int grow = row0 + rt * 16 + hf * 8 + i;
      float val = acc[j][i] + bv;
      if (col < Dn) out0[(size_t)grow * Dn + col] = val;
      else          out1[(size_t)grow * Dn + (col - Dn)] = val;
    }
  }
}

// ---------------------------------------------------------------------------
// Attention: one block = one head x one 16-query tile. Full-row softmax.
// K/V chunks double-buffered through the TDM; V chunk 0 overlaps softmax.
// ---------------------------------------------------------------------------
__global__ __launch_bounds__(256) void attn_kernel(
    const float* __restrict__ Qbuf, const float* __restrict__ Kbuf,
    const float* __restrict__ Vbuf, const unsigned char* __restrict__ amask,
    const int* __restrict__ lengths, float* __restrict__ Abuf) {
  extern __shared__ float sm[];
  float* Qs  = sm;               // 16 x 68
  float* Kb0 = Qs + 16 * 68;     // 128 x 68
  float* Kb1 = Kb0 + 128 * 68;   // 128 x 68
  float* Sc  = Kb1 + 128 * 68;   // 16 x 1156
  float* red = Sc + 16 * 1156;   // 16 x 16
  float* rowmax = red + 256;     // 16
  float* rowsum = rowmax + 16;   // 16
  float* Pp = rowsum + 16;       // 2 x 16 x 64

  const int tid = threadIdx.x;
  const int w = tid >> 5, lane = tid & 31;
  const int hf = lane >> 4, l = lane & 15;
  const int head = blockIdx.y;
  const int bb = head >> 4, hh = head & 15;
  const int q0 = blockIdx.x * 16;
  const int NKC = KLn / 128;     // 9 chunks

  // klengths = lengths + M + R + T - max(lengths)
  int l0 = lengths[0], l1 = lengths[1], l2 = lengths[2], l3 = lengths[3];
  int mx = l0 > l1 ? l0 : l1; mx = mx > l2 ? mx : l2; mx = mx > l3 ? mx : l3;
  int lb = (bb == 0) ? l0 : (bb == 1) ? l1 : (bb == 2) ? l2 : l3;
  const int klen = lb + Mn + Rn + Tn - mx;

#if HAVE_TDM
  if (w == 0) {  // prologue: Q tile + K chunk 0
    tdm_load_2d(Qbuf + ((size_t)q0 * Bn + bb) * Dn + hh * DHn, lds_addr_of(Qs),
                64, 16, Bn * Dn, 5, 3, true);
    tdm_load_2d(Kbuf + ((size_t)bb) * Dn + hh * DHn, lds_addr_of(Kb0),
                64, 128, Bn * Dn, 5, 3, true);
  }
#else
  {
    int row = tid >> 4, c = (tid & 15) << 2;
    float4 v = *(const float4*)(Qbuf + ((size_t)(q0 + row) * Bn + bb) * Dn +
                                hh * DHn + c);
    Qs[row * 68 + c + 0] = v.x; Qs[row * 68 + c + 1] = v.y;
    Qs[row * 68 + c + 2] = v.z; Qs[row * 68 + c + 3] = v.w;
  }
#endif

  // Phase 1: scores S = scale * Q K^T with masks
  for (int ic = 0; ic < NKC; ++ic) {
    const int kc = ic * 128;
    __syncthreads();  // compute of ic-1 done: safe to overwrite other buffer
#if HAVE_TDM
    if (w == 0) {
      if (ic + 1 < NKC) {
        tdm_load_2d(Kbuf + ((size_t)(kc + 128) * Bn + bb) * Dn + hh * DHn,
                    lds_addr_of((ic & 1) ? Kb0 : Kb1), 64, 128, Bn * Dn, 5, 3, true);
        __builtin_amdgcn_s_wait_tensorcnt(1);
      } else {
        __builtin_amdgcn_s_wait_tensorcnt(0);
      }
    }
#else
    {
      float* Kst = (ic & 1) ? Kb1 : Kb0;
#pragma unroll
      for (int j = 0; j < 8; ++j) {
        int vi = tid + j * 256;
        int row = vi >> 4, c = (vi & 15) << 2;
        float4 v = *(const float4*)(Kbuf + ((size_t)(kc + row) * Bn + bb) * Dn +
                                    hh * DHn + c);
        Kst[row * 68 + c + 0] = v.x; Kst[row * 68 + c + 1] = v.y;
        Kst[row * 68 + c + 2] = v.z; Kst[row * 68 + c + 3] = v.w;
      }
    }
#endif
    __syncthreads();

    const float* Kc = (ic & 1) ? Kb1 : Kb0;
    v8f acc = vzero8();
    const int kk0 = kc + w * 16;
#pragma unroll
    for (int d0 = 0; d0 < DHn; d0 += 4) {
      v2f a, b;
      a.x = Qs[l * 68 + d0 + 2 * hf];
      a.y = Qs[l * 68 + d0 + 2 * hf + 1];
      b.x = Kc[(w * 16 + l) * 68 + d0 + 2 * hf];
      b.y = Kc[(w * 16 + l) * 68 + d0 + 2 * hf + 1];
      acc = wmma4(a, b, acc);
    }
#pragma unroll
    for (int i = 0; i < 8; ++i) {
      int m = hf * 8 + i;
      int qg = q0 + m;
      int kg = kk0 + l;
      float val = acc[i] * SCALING;
      if (amask[(size_t)qg * KLn + kg]) val = NEG_INF;
      if (kg >= klen) val = NEG_INF;
      Sc[m * 1156 + kg] = val;
    }
  }
  __syncthreads();

#if HAVE_TDM
  if (w == 0)  // V chunk 0 streams in underneath the softmax phase
    tdm_load_2d(Vbuf + ((size_t)bb) * Dn + hh * DHn, lds_addr_of(Kb0),
                64, 128, Bn * Dn, 5, 3, true);
#endif

  // Phase 2: softmax over 1152 cols, 16 threads per row (exp stored, norm later)
  {
    int row = tid >> 4, t = tid & 15;
    float m = -3.4e38f;
    for (int j = 0; j < KLn / 16; ++j)
      m = fmaxf(m, Sc[row * 1156 + t + 16 * j]);
    red[row * 16 + t] = m;
    __syncthreads();
    if (t == 0) {
      float mm = red[row * 16];
      for (int u = 1; u < 16; ++u) mm = fmaxf(mm, red[row * 16 + u]);
      rowmax[row] = mm;
    }
    __syncthreads();
    float mm = rowmax[row];
    float s = 0.0f;
    for (int j = 0; j < KLn / 16; ++j) {
      int idx = row * 1156 + t + 16 * j;
      float e = __expf(Sc[idx] - mm);
      Sc[idx] = e;
      s += e;
    }
    red[row * 16 + t] = s;
    __syncthreads();
    if (t == 0) {
      float ss = 0.0f;
      for (int u = 0; u < 16; ++u) ss += red[row * 16 + u];
      rowsum[row] = ss;
    }
  }

  // Phase 3: attn = P @ V ; waves: 4 col tiles x 2 K-subranges per 128-chunk
  {
    const int ct = w & 3, d0c = ct * 16;
    const int hk = w >> 2;
    v8f acc = vzero8();
    for (int ic = 0; ic < NKC; ++ic) {
      const int kc = ic * 128;
      __syncthreads();
#if HAVE_TDM
      if (w == 0) {
        if (ic + 1 < NKC) {
          tdm_load_2d(Vbuf + ((size_t)(kc + 128) * Bn + bb) * Dn + hh * DHn,
                      lds_addr_of((ic & 1) ? Kb0 : Kb1), 64, 128, Bn * Dn, 5, 3, true);
          __builtin_amdgcn_s_wait_tensorcnt(1);
        } else {
          __builtin_amdgcn_s_wait_tensorcnt(0);
        }
      }
#else
      {
        float* Vst = (ic & 1) ? Kb1 : Kb0;
#pragma unroll
        for (int j = 0; j < 8; ++j) {
          int vi = tid + j * 256;
          int row = vi >> 4, c = (vi & 15) << 2;
          float4 v = *(const float4*)(Vbuf + ((size_t)(kc + row) * Bn + bb) * Dn +
                                      hh * DHn + c);
          Vst[row * 68 + c + 0] = v.x; Vst[row * 68 + c + 1] = v.y;
          Vst[row * 68 + c + 2] = v.z; Vst[row * 68 + c + 3] = v.w;
        }
      }
#endif
      __syncthreads();

      const float* Vc = (ic & 1) ? Kb1 : Kb0;
      const int kb = hk * 64;
#pragma unroll 4
      for (int kk = 0; kk < 64; kk += 4) {
        int ks = kb + kk;
        v2f a, b;
        a.x = Sc[l * 1156 + kc + ks + 2 * hf];
        a.y = Sc[l * 1156 + kc + ks + 2 * hf + 1];
        b.x = Vc[(ks + 2 * hf) * 68 + d0c + l];
        b.y = Vc[(ks + 2 * hf + 1) * 68 + d0c + l];
        acc = wmma4(a, b, acc);
      }
    }
#pragma unroll
    for (int i = 0; i < 8; ++i) {
      int m = hf * 8 + i;
      Pp[hk * 1024 + m * 64 + d0c + l] = acc[i];
    }
  }
  __syncthreads();

  // Combine halves, normalize, store to Abuf (layout (Q, B, D))
  {
    int idx = tid * 4;
    int row = idx >> 6, c = idx & 63;
    float inv = 1.0f / rowsum[row];
    float* dst = Abuf + ((size_t)(q0 + row) * Bn + bb) * Dn + hh * DHn + c;
#pragma unroll
    for (int u = 0; u < 4; ++u)
      dst[u] = (Pp[row * 64 + c + u] + Pp[1024 + row * 64 + c + u]) * inv;
  }
}

// ---------------------------------------------------------------------------
// Output projection + split store. X and W tiles both TDM, double-buffered.
// ---------------------------------------------------------------------------
__global__ __launch_bounds__(256) void out_gemm(
    const float* __restrict__ A, const float* __restrict__ W,
    const float* __restrict__ bias, float* __restrict__ dout) {
  extern __shared__ float sm[];
  float* Xb0 = sm;               // 64 x 68
  float* Xb1 = Xb0 + 64 * 68;
  float* Wb0 = Xb1 + 64 * 68;    // 64 x 132
  float* Wb1 = Wb0 + 64 * 132;

  const int tid = threadIdx.x;
  const int w = tid >> 5, lane = tid & 31;
  const int hf = lane >> 4, l = lane & 15;
  const int rt = w >> 1, cg = w & 1;
  const int row0 = blockIdx.x * 64;
  const int nbase = blockIdx.y * 128;
  const int NCH = Dn / 64;

  v8f acc[4];
#pragma unroll
  for (int j = 0; j < 4; ++j) acc[j] = vzero8();

#if HAVE_TDM
  if (w == 0) {
    tdm_load_2d(A + (size_t)row0 * Dn, lds_addr_of(Xb0), 64, 64, Dn, 5, 3, true);
    tdm_load_2d(W + nbase, lds_addr_of(Wb0), 128, 64, Dn, 6, 3, true);
  }
#endif

  for (int ic = 0; ic < NCH; ++ic) {
    const int k0 = ic * 64;
    __syncthreads();
#if HAVE_TDM
    if (w == 0) {
      if (ic + 1 < NCH) {
        tdm_load_2d(A + (size_t)row0 * Dn + k0 + 64,
                    lds_addr_of((ic & 1) ? Xb0 : Xb1), 64, 64, Dn, 5, 3, true);
        tdm_load_2d(W + (size_t)(k0 + 64) * Dn + nbase,
                    lds_addr_of((ic & 1) ? Wb0 : Wb1), 128, 64, Dn, 6, 3, true);
        __builtin_amdgcn_s_wait_tensorcnt(2);  // current pair landed
      } else {
        __builtin_amdgcn_s_wait_tensorcnt(0);
      }
    }
#else
    {
      float* Xst = (ic & 1) ? Xb1 : Xb0;
      float* Wst = (ic & 1) ? Wb1 : Wb0;
#pragma unroll
      for (int j = 0; j < 4; ++j) {
        int vi = tid + j * 256;
        int row = vi >> 4, c = (vi & 15) << 2;
        float4 v = *(const float4*)(A + (size_t)(row0 + row) * Dn + k0 + c);
        Xst[row * 68 + c + 0] = v.x; Xst[row * 68 + c + 1] = v.y;
        Xst[row * 68 + c + 2] = v.z; Xst[row * 68 + c + 3] = v.w;
      }
#pragma unroll
      for (int j = 0; j < 8; ++j) {
        int vi = tid + j * 256;
        int row = vi >> 5, c = (vi & 31) << 2;
        float4 v = *(const float4*)(W + (size_t)(k0 + row) * Dn + nbase + c);
        Wst[row * 132 + c + 0] = v.x; Wst[row * 132 + c + 1] = v.y;
        Wst[row * 132 + c + 2] = v.z; Wst[row * 132 + c + 3] = v.w;
      }
    }
#endif
    __syncthreads();

    const float* Xc = (ic & 1) ? Xb1 : Xb0;
    const float* Wc = (ic & 1) ? Wb1 : Wb0;
#pragma unroll 4
    for (int kk = 0; kk < 64; kk += 4) {
      v2f a;
      a.x = Xc[(rt * 16 + l) * 68 + kk + 2 * hf];
      a.y = Xc[(rt * 16 + l) * 68 + kk + 2 * hf + 1];
#pragma unroll
      for (int j = 0; j < 4; ++j) {
        int nt = cg * 64 + j * 16;
        v2f b;
        b.x = Wc[(kk + 2 * hf) * 132 + nt + l];
        b.y = Wc[(kk + 2 * hf + 1) * 132 + nt + l];
        acc[j] = wmma4(a, b, acc[j]);
      }
    }
  }

#pragma unroll
  for (int j = 0; j < 4; ++j) {
    int col = nbase + cg * 64 + j * 16 + l;
    float bv = bias[col];
#pragma unroll
    for (int i = 0; i < 8; ++i) {
      int grow = row0 + rt * 16 + hf * 8 + i;   // grow = q*B + b
      int q = grow / Bn;
      float val = acc[j][i] + bv;
      if (q < QSn) {
        dout[(size_t)grow * Dn + col] = val;
      } else {
        int bbx = grow - q * Bn;
        size_t off = (size_t)QSn * Bn * Dn +
                     ((size_t)(q - QSn) * Bn + bbx) * Dn + col;
        dout[off] = tanhf(val);
      }
    }
  }
}

extern "C" void kernel_launch(void* const* d_in, const int* in_sizes, int n_in,
                              void* d_out, int out_size, void* d_ws, size_t ws_size,
                              hipStream_t stream) {
  const float* utt  = (const float*)d_in[0];
  const float* rc   = (const float*)d_in[1];
  const float* summ = (const float*)d_in[2];
  const float* mems = (const float*)d_in[3];
  const int*   lengths = (const int*)d_in[4];
  const unsigned char* amask = (const unsigned char*)d_in[5];
  const float* Wq   = (const float*)d_in[6];
  const float* bq   = (const float*)d_in[7];
  const float* Wkv  = (const float*)d_in[8];
  const float* bkv  = (const float*)d_in[9];
  const float* Wout = (const float*)d_in[10];
  const float* bout = (const float*)d_in[11];
  float* out = (float*)d_out;

  // Workspace layout (floats): Qbuf | Kbuf | Vbuf | Abuf (~74 MB)
  float* Qbuf = (float*)d_ws;
  float* Kb = Qbuf + (size_t)QBn * Dn;
  float* Vb = Kb + (size_t)KLBn * Dn;
  float* Ab = Vb + (size_t)KLBn * Dn;

  size_t smemProj = (size_t)(64 * 68 + 2 * 64 * 132) * sizeof(float);
  size_t smemAttn = (size_t)(16 * 68 + 2 * 128 * 68 + 16 * 1156 + 256 + 16 + 16 +
                             2048) * sizeof(float);
  size_t smemOut  = (size_t)(2 * 64 * 68 + 2 * 64 * 132) * sizeof(float);

  proj_gemm<<<dim3(QBn / 64, Dn / 128), 256, smemProj, stream>>>(
      rc, Rn * Bn, utt, Tn * Bn, summ, Wq, bq, Dn, Qbuf, Qbuf);

  proj_gemm<<<dim3(KLBn / 64, (2 * Dn) / 128), 256, smemProj, stream>>>(
      mems, Mn * Bn, rc, Rn * Bn, utt, Wkv, bkv, 2 * Dn, Kb, Vb);

  attn_kernel<<<dim3(Qn / 16, Bn * Hn), 256, smemAttn, stream>>>(
      Qbuf, Kb, Vb, amask, lengths, Ab);

  out_gemm<<<dim3(QBn / 64, Dn / 128), 256, smemOut, stream>>>(Ab, Wout, bout, out);

  size_t keyOff = (size_t)(QSn + Sn) * Bn * Dn;
  size_t kvLen  = (size_t)Tn * Bn * Dn;
  size_t srcOff = (size_t)(Mn + Rn) * Bn * Dn;
  hipMemcpyAsync(out + keyOff, Kb + srcOff, kvLen * sizeof(float),
                 hipMemcpyDeviceToDevice, stream);
  hipMemcpyAsync(out + keyOff + kvLen, Vb + srcOff, kvLen * sizeof(float),
                 hipMemcpyDeviceToDevice, stream);
}